// GraphSageLayer_62569083568400
// MI455X (gfx1250) — compile-verified
//
#include <hip/hip_runtime.h>

#define N_NODES 10000
#define IN_FEAT 512
#define OUT_FEAT 512
#define N_EDGES 160000

typedef __attribute__((ext_vector_type(2))) float v2f;
typedef __attribute__((ext_vector_type(8))) float v8f;
// GCC-style vector exactly matching the async-to-LDS builtin's pointee type
// (clang diagnostic: "__attribute__((__vector_size__(4 * sizeof(int)))) int __device__ *").
typedef int v4i_vs __attribute__((vector_size(4 * sizeof(int))));

#if __has_builtin(__builtin_amdgcn_global_load_async_to_lds_b128) && \
    __has_builtin(__builtin_amdgcn_s_wait_asynccnt)
#define HAVE_ASYNC_LDS 1
#else
#define HAVE_ASYNC_LDS 0
#endif

// Cast helpers: integer round-trip sidesteps const / addrspace conversion
// rules; flat->LDS is a 32-bit truncation on amdgcn (flat shared addresses
// keep the LDS byte offset in the low 32 bits), matching addrspacecast.
#define AS1V(p) ((__attribute__((address_space(1))) v4i_vs*)(uintptr_t)(p))
#define AS3V(p) ((__attribute__((address_space(3))) v4i_vs*)(uintptr_t)(p))

// ---------------------------------------------------------------------------
// K0: zero the per-node edge histogram.
// ---------------------------------------------------------------------------
__global__ __launch_bounds__(256) void sage_zero(int* __restrict__ count) {
  const int i = blockIdx.x * blockDim.x + threadIdx.x;
  if (i < N_NODES) count[i] = 0;
}

// ---------------------------------------------------------------------------
// K1: histogram dst degrees (160K int atomics -- cheap, L2 resident).
// ---------------------------------------------------------------------------
__global__ __launch_bounds__(256) void sage_count(const int* __restrict__ dst,
                                                  int* __restrict__ count) {
  const int e = blockIdx.x * blockDim.x + threadIdx.x;
  if (e < N_EDGES) atomicAdd(&count[dst[e]], 1);
}

// ---------------------------------------------------------------------------
// K2: single-block exclusive scan over 10000 counts -> CSR row starts.
// 256 threads x 40 nodes each; 256-wide partial scan done serially by t0
// (256 adds -- negligible). Also primes the bucket-fill cursors.
// ---------------------------------------------------------------------------
__global__ __launch_bounds__(256) void sage_scan(const int* __restrict__ count,
                                                 int* __restrict__ start,
                                                 int* __restrict__ cursor) {
  __shared__ int part[256];
  const int t = threadIdx.x;
  const int chunk = (N_NODES + 255) / 256;  // 40
  const int lo = t * chunk;
  const int hi = (lo + chunk < N_NODES) ? lo + chunk : N_NODES;

  int s = 0;
  for (int i = lo; i < hi; ++i) s += count[i];
  part[t] = s;
  __syncthreads();

  if (t == 0) {
    int acc = 0;
    for (int i = 0; i < 256; ++i) { int v = part[i]; part[i] = acc; acc += v; }
  }
  __syncthreads();

  int acc = part[t];
  for (int i = lo; i < hi; ++i) {
    start[i] = acc;
    cursor[i] = acc;
    acc += count[i];
  }
  if (t == 0) start[N_NODES] = N_EDGES;
}

// ---------------------------------------------------------------------------
// K3: bucket-fill edge sources into CSR order (160K int atomics).
// ---------------------------------------------------------------------------
__global__ __launch_bounds__(256) void sage_bucket(const int* __restrict__ src,
                                                   const int* __restrict__ dst,
                                                   int* __restrict__ cursor,
                                                   int* __restrict__ order) {
  const int e = blockIdx.x * blockDim.x + threadIdx.x;
  if (e < N_EDGES) {
    const int pos = atomicAdd(&cursor[dst[e]], 1);
    order[pos] = src[e];
  }
}

// ---------------------------------------------------------------------------
// K4: wave-per-node gather-reduce. Replaces 82M f32 atomics with pure
// streaming float4 loads (x is L2-resident) + register accumulation.
// Lane covers features {4*lane, 4*lane+128, +256, +384} as float4s.
// ---------------------------------------------------------------------------
__global__ __launch_bounds__(256) void sage_reduce(const float* __restrict__ x,
                                                   const int* __restrict__ start,
                                                   const int* __restrict__ order,
                                                   float* __restrict__ summed) {
  const int lane = threadIdx.x & 31;
  const int n = blockIdx.x * (blockDim.x >> 5) + (threadIdx.x >> 5);
  if (n >= N_NODES) return;

  const int base = start[n];
  const int cnt = start[n + 1] - base;

  float4 a0 = {0, 0, 0, 0}, a1 = {0, 0, 0, 0}, a2 = {0, 0, 0, 0}, a3 = {0, 0, 0, 0};
  for (int j = 0; j < cnt; ++j) {
    const int s = order[base + j];
    const float4* __restrict__ xs = (const float4*)(x + (size_t)s * IN_FEAT);
    const float4 v0 = xs[lane];
    const float4 v1 = xs[lane + 32];
    const float4 v2 = xs[lane + 64];
    const float4 v3 = xs[lane + 96];
    a0.x += v0.x; a0.y += v0.y; a0.z += v0.z; a0.w += v0.w;
    a1.x += v1.x; a1.y += v1.y; a1.z += v1.z; a1.w += v1.w;
    a2.x += v2.x; a2.y += v2.y; a2.z += v2.z; a2.w += v2.w;
    a3.x += v3.x; a3.y += v3.y; a3.z += v3.z; a3.w += v3.w;
  }
  float4* __restrict__ sm = (float4*)(summed + (size_t)n * IN_FEAT);
  sm[lane] = a0;
  sm[lane + 32] = a1;
  sm[lane + 64] = a2;
  sm[lane + 96] = a3;
}

// ---------------------------------------------------------------------------
// K5: fused mean-normalize + dual GEMM + ReLU via V_WMMA_F32_16X16X4_F32.
// Block = 8 waves = 16 rows x 128 out-cols. The shared 16x512 x-tile and
// mean-tile are staged in LDS (async-to-LDS when available), rows padded
// to 516 floats so the A-fragment b64 reads are LDS-bank-conflict-free:
// bank = (4*row + col) mod 64 covers all 64 banks across 32 lanes.
// A-frag: lane L -> row (L&15), k pair k+2*(L>>4)   (ds_load_b64)
// B-frag: lane L -> w row o0+(L&15), same k pair    (global b64, L2-resident)
// C/D:    vgpr v -> row v + 8*(L>>4), col (L&15)
// ---------------------------------------------------------------------------
#define LDS_STRIDE 516

__global__ __launch_bounds__(256) void sage_gemm(const float* __restrict__ x,
                                                 const float* __restrict__ summed,
                                                 const int* __restrict__ start,
                                                 const float* __restrict__ w,
                                                 float* __restrict__ out) {
  __shared__ float lsA[16 * LDS_STRIDE];
  __shared__ float lsM[16 * LDS_STRIDE];

  const int tid = threadIdx.x;
  const int lane = tid & 31;
  const int wave = tid >> 5;
  const int n0 = (blockIdx.x >> 2) * 16;        // 625 row bands (exact)
  const int o0 = (blockIdx.x & 3) * 128 + wave * 16;

  // Cooperative tile fill: 16 rows x 128 float4 per array, 8 iters/thread.
  for (int idx = tid; idx < 16 * 128; idx += 256) {
    const int row = idx >> 7;
    const int c4 = idx & 127;
    const size_t goff = (size_t)(n0 + row) * IN_FEAT + c4 * 4;
    const int loff = row * LDS_STRIDE + c4 * 4;
#if HAVE_ASYNC_LDS
    __builtin_amdgcn_global_load_async_to_lds_b128(AS1V(x + goff), AS3V(&lsA[loff]), 0, 0);
    __builtin_amdgcn_global_load_async_to_lds_b128(AS1V(summed + goff), AS3V(&lsM[loff]), 0, 0);
#else
    *(float4*)(&lsA[loff]) = *(const float4*)(x + goff);
    *(float4*)(&lsM[loff]) = *(const float4*)(summed + goff);
#endif
  }
#if HAVE_ASYNC_LDS
  __builtin_amdgcn_s_wait_asynccnt(0);
#endif
  __syncthreads();

  const int r = lane & 15;   // A row / B col (w row)
  const int kh = lane >> 4;  // K-pair half owned by this lane

  // Fold the mean into the A-fragment read: one v_rcp_f32 per tile.
  const int cnt = start[n0 + r + 1] - start[n0 + r];
  const float degf = (cnt > 1) ? (float)cnt : 1.0f;
#if __has_builtin(__builtin_amdgcn_rcpf)
  const float inv = __builtin_amdgcn_rcpf(degf);
#else
  const float inv = 1.0f / degf;
#endif

  const float* __restrict__ aL = lsA + r * LDS_STRIDE + 2 * kh;
  const float* __restrict__ mL = lsM + r * LDS_STRIDE + 2 * kh;
  const float* __restrict__ brow = w + (size_t)(o0 + r) * (2 * IN_FEAT) + 2 * kh;

  v8f c = {};

  // Self term: x . W1
  #pragma unroll 8
  for (int k = 0; k < IN_FEAT; k += 4) {
    v2f a = *(const v2f*)(aL + k);
    v2f b = *(const v2f*)(brow + k);
    c = __builtin_amdgcn_wmma_f32_16x16x4_f32(false, a, false, b, (short)0, c, false, false);
  }

  // Neighbor-mean term: (summed * invdeg) . W2
  #pragma unroll 8
  for (int k = 0; k < IN_FEAT; k += 4) {
    v2f a = *(const v2f*)(mL + k);
    a *= inv;
    v2f b = *(const v2f*)(brow + IN_FEAT + k);
    c = __builtin_amdgcn_wmma_f32_16x16x4_f32(false, a, false, b, (short)0, c, false, false);
  }

  // Writeback with ReLU.
  #pragma unroll
  for (int v = 0; v < 8; ++v) {
    const int row = n0 + v + 8 * kh;
    const float val = c[v];
    out[(size_t)row * OUT_FEAT + (o0 + r)] = val > 0.0f ? val : 0.0f;
  }
}

// ---------------------------------------------------------------------------
// Launcher (graph-capture safe: only kernel launches on `stream`).
// Workspace layout (~21.3 MB):
//   summed : float[N*F]      @ 0
//   count  : int[N]          @ 20,480,000
//   start  : int[N+1]        @ 20,520,000
//   cursor : int[N]          @ 20,560,004
//   order  : int[E]          @ 20,600,004
// ---------------------------------------------------------------------------
extern "C" void kernel_launch(void* const* d_in, const int* in_sizes, int n_in,
                              void* d_out, int out_size, void* d_ws, size_t ws_size,
                              hipStream_t stream) {
  const float* x = (const float*)d_in[0];  // [N, F]
  const float* w = (const float*)d_in[1];  // [OUT, 2F]
  const int* ei = (const int*)d_in[2];     // [2, E]
  const int* src = ei;
  const int* dst = ei + N_EDGES;
  float* out = (float*)d_out;              // [N, OUT]

  char* ws = (char*)d_ws;
  float* summed = (float*)ws;
  int* count = (int*)(ws + (size_t)N_NODES * IN_FEAT * 4);
  int* start = count + N_NODES;
  int* cursor = start + (N_NODES + 1);
  int* order = cursor + N_NODES;

  (void)in_sizes; (void)n_in; (void)out_size; (void)ws_size;

  sage_zero<<<(N_NODES + 255) / 256, 256, 0, stream>>>(count);
  sage_count<<<(N_EDGES + 255) / 256, 256, 0, stream>>>(dst, count);
  sage_scan<<<1, 256, 0, stream>>>(count, start, cursor);
  sage_bucket<<<(N_EDGES + 255) / 256, 256, 0, stream>>>(src, dst, cursor, order);
  sage_reduce<<<(N_NODES + 7) / 8, 256, 0, stream>>>(x, start, order, summed);

  const int blocks = (N_NODES / 16) * (OUT_FEAT / 128);  // 625 * 4 = 2500 (exact)
  sage_gemm<<<blocks, 256, 0, stream>>>(x, summed, start, w, out);
}